// EulerIntegration_1984274890836
// MI455X (gfx1250) — compile-verified
//
#include <hip/hip_runtime.h>
#include <hip/hip_bf16.h>

// Euler motion-field integration, MI455X (gfx1250, wave32).
// H=1024, W=1920 fixed by the reference setup_inputs().
// Strategy:
//   1) pack_kernel: repack planar motion [2,H,W] -> interleaved float2 in d_ws
//      (streaming b64 loads -> b128 stores; also warms the 192MB L2 with the
//      whole 15.7MB field).
//   2) euler_packed: per-pixel serial chain of `steps` gathers, each a single
//      global_load_b64 from L2-resident packed field. Sticky-invalid lanes
//      break early (EXEC-masked). NT stores for the write-once output.
// Fallback euler_planar used only if ws_size is too small for the packed field.

#define EH 1024
#define EW 1920
#define EHW (EH * EW)
#define OOB_FILL 1921.0f

__global__ __launch_bounds__(256) void pack_kernel(const float* __restrict__ motion,
                                                   float4* __restrict__ packed) {
    int t = blockIdx.x * blockDim.x + threadIdx.x;      // handles pixels 2t, 2t+1
    if (t >= EHW / 2) return;
    const float2* mx = (const float2*)(motion);          // x-flow plane
    const float2* my = (const float2*)(motion + EHW);    // y-flow plane
    float2 a = mx[t];
    float2 b = my[t];
    // packed as float2 pairs: pixel 2t -> (a.x, b.x), pixel 2t+1 -> (a.y, b.y)
    packed[t] = make_float4(a.x, b.x, a.y, b.y);
}

__global__ __launch_bounds__(256) void euler_packed(const float2* __restrict__ mot,
                                                    const int* __restrict__ dframe,
                                                    float* __restrict__ out) {
    int idx = blockIdx.x * blockDim.x + threadIdx.x;
    if (idx >= EHW) return;

    const int x0i = idx % EW;
    const int y0i = idx / EW;
    const float x0 = (float)x0i;
    const float y0 = (float)y0i;

    const int steps = *dframe;

    float dx = x0;
    float dy = y0;
    bool invalid = false;

    for (int t = 0; t < steps; ++t) {
        // round-half-even matches jnp.round; positions are provably in-bounds here
        int xi = __float2int_rn(dx);
        int yi = __float2int_rn(dy);
        float2 m = mot[yi * EW + xi];          // single global_load_b64 on the chain
        dx += m.x;
        dy += m.y;
        if (dx > (float)(EW - 1) || dx < 0.0f || dy > (float)(EH - 1) || dy < 0.0f) {
            invalid = true;                     // sticky: output frozen at OOB_FILL
            break;                              // lane goes EXEC-inactive, stops gathering
        }
    }

    float ox, oy;
    if (invalid) {
        ox = OOB_FILL;
        oy = OOB_FILL;
    } else {
        ox = dx - x0;
        oy = dy - y0;
    }
    // write-once output: non-temporal so it doesn't evict the L2-resident field
    __builtin_nontemporal_store(ox, out + idx);
    __builtin_nontemporal_store(oy, out + EHW + idx);
}

// Fallback: gather from the two planes separately (2x global_load_b32 per step).
__global__ __launch_bounds__(256) void euler_planar(const float* __restrict__ motion,
                                                    const int* __restrict__ dframe,
                                                    float* __restrict__ out) {
    int idx = blockIdx.x * blockDim.x + threadIdx.x;
    if (idx >= EHW) return;

    const int x0i = idx % EW;
    const int y0i = idx / EW;
    const float x0 = (float)x0i;
    const float y0 = (float)y0i;

    const int steps = *dframe;
    const float* __restrict__ motx = motion;
    const float* __restrict__ moty = motion + EHW;

    float dx = x0;
    float dy = y0;
    bool invalid = false;

    for (int t = 0; t < steps; ++t) {
        int xi = __float2int_rn(dx);
        int yi = __float2int_rn(dy);
        int g = yi * EW + xi;
        float mx = motx[g];
        float my = moty[g];
        dx += mx;
        dy += my;
        if (dx > (float)(EW - 1) || dx < 0.0f || dy > (float)(EH - 1) || dy < 0.0f) {
            invalid = true;
            break;
        }
    }

    float ox, oy;
    if (invalid) {
        ox = OOB_FILL;
        oy = OOB_FILL;
    } else {
        ox = dx - x0;
        oy = dy - y0;
    }
    __builtin_nontemporal_store(ox, out + idx);
    __builtin_nontemporal_store(oy, out + EHW + idx);
}

extern "C" void kernel_launch(void* const* d_in, const int* in_sizes, int n_in,
                              void* d_out, int out_size, void* d_ws, size_t ws_size,
                              hipStream_t stream) {
    const float* motion = (const float*)d_in[0];   // [1,2,H,W] float32
    const int* dframe   = (const int*)d_in[1];     // scalar int (60)
    float* out          = (float*)d_out;           // [1,2,H,W] float32

    const size_t packed_bytes = (size_t)EHW * 2 * sizeof(float);   // 15.7 MB

    const int threads = 256;                       // 8 wave32 waves per block
    const int grid_main = (EHW + threads - 1) / threads;

    if (d_ws != nullptr && ws_size >= packed_bytes) {
        const int grid_pack = (EHW / 2 + threads - 1) / threads;
        pack_kernel<<<grid_pack, threads, 0, stream>>>(motion, (float4*)d_ws);
        euler_packed<<<grid_main, threads, 0, stream>>>((const float2*)d_ws, dframe, out);
    } else {
        euler_planar<<<grid_main, threads, 0, stream>>>(motion, dframe, out);
    }
}